// AdaptiveInput_18786186953331
// MI455X (gfx1250) — compile-verified
//
#include <hip/hip_runtime.h>
#include <hip/hip_bf16.h>

// ---------------------------------------------------------------------------
// AdaptiveInput forward for MI455X (gfx1250, wave32).
// Strategy: classify tokens into clusters, compact tail-cluster token lists,
// then run fp32 WMMA (V_WMMA_F32_16X16X4_F32) batched GEMMs:
//   out[16 tokens x 64 feats] per wave, K stepped by 4.
// fp32 WMMA keeps reference precision; the problem is HBM-bound (~180MB
// traffic vs ~20 GFLOPs), so f32 matrix ops are the right precision choice.
// ---------------------------------------------------------------------------

typedef float v2f __attribute__((ext_vector_type(2)));
typedef float v8f __attribute__((ext_vector_type(8)));

#define CUT1 5000
#define CUT2 20000
#define IN_FEAT 1024

__global__ void ai_init_counters(int* cnt) {
    if (threadIdx.x < 2) cnt[threadIdx.x] = 0;
}

__global__ void ai_classify(const int* __restrict__ tokens, int N,
                            int* __restrict__ cnt,
                            int* __restrict__ idx0, int* __restrict__ idx1) {
    int n = blockIdx.x * blockDim.x + threadIdx.x;
    if (n >= N) return;
    int t = tokens[n];
    if (t >= CUT1 && t < CUT2) {
        int p = atomicAdd(&cnt[0], 1);
        idx0[p] = n;
    } else if (t >= CUT2) {
        int p = atomicAdd(&cnt[1], 1);
        idx1[p] = n;
    }
}

// Head cluster: straight row gather, float4 (global_load_b128 / store_b128).
__global__ void ai_head_copy(const int* __restrict__ tokens,
                             const float* __restrict__ head,
                             float* __restrict__ out) {
    int n = blockIdx.x;
    int t = tokens[n];
    if (t >= CUT1) return;
    const float4* src = (const float4*)(head + (size_t)t * IN_FEAT);
    float4* dst = (float4*)(out + (size_t)n * IN_FEAT);
    dst[threadIdx.x] = src[threadIdx.x];   // blockDim.x == 256 (= 1024/4)
}

// Tail projection GEMM: one wave computes a 16x64 output tile.
//   A (16x4 f32):  lane L holds rows m = L&15, K = 2*(L>>4)+{0,1}
//   B (4x16 f32):  lane L holds col n = L&15, K = 2*(L>>4)+{0,1}
//   C (16x16 f32): VGPR j: lanes 0-15 -> M=j, lanes 16-31 -> M=j+8, N=L&15
template <int H>
__global__ __launch_bounds__(32) void ai_tail_gemm(
    const int* __restrict__ tokens,
    const int* __restrict__ idx,       // compacted token positions
    const int* __restrict__ cnt_p,     // device-side count for this cluster
    const float* __restrict__ table,   // [rows, H]
    const float* __restrict__ W,       // [1024, H] row-major (torch Linear)
    float* __restrict__ out,           // [N, 1024]
    int low) {
    int count = *cnt_p;
    int mtile = blockIdx.x;
    if (mtile * 16 >= count) return;   // wave-uniform: EXEC stays all-1s

    int lane = threadIdx.x;
    int half = lane >> 4;              // selects K pair {0,1} vs {2,3}
    int mn   = lane & 15;              // row (for A) / col (for B,C)
    int colbase = blockIdx.y * 64;

    // A operand: this lane's token row in the tail table.
    int slotA = mtile * 16 + mn;
    if (slotA >= count) slotA = count - 1;       // ragged tile: clamp
    int tokA = idx[slotA];
    int rowA = tokens[tokA] - low;
    const float* aptr = table + (size_t)rowA * H + half * 2;

    // B operands: four 16-wide feature tiles (cols colbase + nt*16 + mn).
    const float* bptr0 = W + (size_t)(colbase + mn +  0) * H + half * 2;
    const float* bptr1 = W + (size_t)(colbase + mn + 16) * H + half * 2;
    const float* bptr2 = W + (size_t)(colbase + mn + 32) * H + half * 2;
    const float* bptr3 = W + (size_t)(colbase + mn + 48) * H + half * 2;

    v8f c0 = {}, c1 = {}, c2 = {}, c3 = {};

#pragma unroll 2
    for (int k = 0; k < H; k += 4) {
        v2f a  = *(const v2f*)(aptr  + k);
        v2f b0 = *(const v2f*)(bptr0 + k);
        v2f b1 = *(const v2f*)(bptr1 + k);
        v2f b2 = *(const v2f*)(bptr2 + k);
        v2f b3 = *(const v2f*)(bptr3 + k);
        c0 = __builtin_amdgcn_wmma_f32_16x16x4_f32(false, a, false, b0,
                                                   (short)0, c0, false, false);
        c1 = __builtin_amdgcn_wmma_f32_16x16x4_f32(false, a, false, b1,
                                                   (short)0, c1, false, false);
        c2 = __builtin_amdgcn_wmma_f32_16x16x4_f32(false, a, false, b2,
                                                   (short)0, c2, false, false);
        c3 = __builtin_amdgcn_wmma_f32_16x16x4_f32(false, a, false, b3,
                                                   (short)0, c3, false, false);
    }

    // Epilogue: scatter rows back to original token positions.
    int mbase = half * 8;
#pragma unroll
    for (int j = 0; j < 8; ++j) {
        int slot = mtile * 16 + mbase + j;
        if (slot >= count) slot = count - 1;     // duplicate writes, same value
        int tok = idx[slot];
        float* orow = out + (size_t)tok * IN_FEAT + colbase + mn;
        orow[0]  = c0[j];
        orow[16] = c1[j];
        orow[32] = c2[j];
        orow[48] = c3[j];
    }
}

extern "C" void kernel_launch(void* const* d_in, const int* in_sizes, int n_in,
                              void* d_out, int out_size, void* d_ws, size_t ws_size,
                              hipStream_t stream) {
    const int*   tokens    = (const int*)d_in[0];
    const float* head_emb  = (const float*)d_in[1];
    const float* tail0_emb = (const float*)d_in[2];
    const float* tail0_w   = (const float*)d_in[3];
    const float* tail1_emb = (const float*)d_in[4];
    const float* tail1_w   = (const float*)d_in[5];
    float* out = (float*)d_out;

    int N = in_sizes[0];               // 8*4096 = 32768 tokens

    int* cnt  = (int*)d_ws;            // [0]=tail0 count, [1]=tail1 count
    int* idx0 = cnt + 8;               // 32B-aligned lists
    int* idx1 = idx0 + N;

    ai_init_counters<<<1, 32, 0, stream>>>(cnt);
    ai_classify<<<(N + 255) / 256, 256, 0, stream>>>(tokens, N, cnt, idx0, idx1);
    ai_head_copy<<<N, IN_FEAT / 4, 0, stream>>>(tokens, head_emb, out);

    dim3 gemm_grid((N + 15) / 16, IN_FEAT / 64);   // worst-case; blocks self-gate
    ai_tail_gemm<512><<<gemm_grid, 32, 0, stream>>>(
        tokens, idx0, &cnt[0], tail0_emb, tail0_w, out, CUT1);
    ai_tail_gemm<256><<<gemm_grid, 32, 0, stream>>>(
        tokens, idx1, &cnt[1], tail1_emb, tail1_w, out, CUT2);
}